// OutputGenerator_56418690400584
// MI455X (gfx1250) — compile-verified
//
#include <hip/hip_runtime.h>
#include <hip/hip_bf16.h>

#define BB 256
#define VV 128000
#define NBINS 2048
#define HLO (-16.0f)
#define BINSCALE 64.0f        // 1/binwidth, range [-16,16) over 2048 bins
#define MAXC 1024
#define NMASKW (VV / 32)      // 4000 u32 words = 16 KB bitmask
#define INV_T (1.0f/0.7f)
#define INV_REP (1.0f/1.2f)
#define TOPP 0.9f

typedef float __attribute__((ext_vector_type(4))) f4;

// ---------------- CDNA5 async global->LDS path (gfx1250, probe-verified) ----
__device__ __forceinline__ void async_ld_b128(unsigned lds_off, unsigned long long gaddr) {
    // VDST = LDS byte address, ADDR = 64-bit global address, saddr=off
    asm volatile("global_load_async_to_lds_b128 %0, %1, off"
                 :: "v"(lds_off), "v"(gaddr) : "memory");
}
__device__ __forceinline__ void wait_async_1() { asm volatile("s_wait_asynccnt 1" ::: "memory"); }
__device__ __forceinline__ void wait_async_0() { asm volatile("s_wait_asynccnt 0" ::: "memory"); }

// ---------------- threefry2x32 (JAX PRNG, key(42) == {0,42}) ----------------
__device__ __forceinline__ unsigned rotl32(unsigned x, int d) { return (x << d) | (x >> (32 - d)); }

__device__ void threefry2x32(unsigned k0, unsigned k1, unsigned x0, unsigned x1,
                             unsigned* o0, unsigned* o1) {
    const unsigned ks2 = k0 ^ k1 ^ 0x1BD11BDAu;
    unsigned X0 = x0 + k0, X1 = x1 + k1;
    #pragma unroll
    for (int i = 0; i < 4; i++) { const int R[4] = {13,15,26,6};  X0 += X1; X1 = rotl32(X1, R[i]); X1 ^= X0; }
    X0 += k1; X1 += ks2 + 1u;
    #pragma unroll
    for (int i = 0; i < 4; i++) { const int R[4] = {17,29,16,24}; X0 += X1; X1 = rotl32(X1, R[i]); X1 ^= X0; }
    X0 += ks2; X1 += k0 + 2u;
    #pragma unroll
    for (int i = 0; i < 4; i++) { const int R[4] = {13,15,26,6};  X0 += X1; X1 = rotl32(X1, R[i]); X1 ^= X0; }
    X0 += k0; X1 += k1 + 3u;
    #pragma unroll
    for (int i = 0; i < 4; i++) { const int R[4] = {17,29,16,24}; X0 += X1; X1 = rotl32(X1, R[i]); X1 ^= X0; }
    X0 += k1; X1 += ks2 + 4u;
    #pragma unroll
    for (int i = 0; i < 4; i++) { const int R[4] = {13,15,26,6};  X0 += X1; X1 = rotl32(X1, R[i]); X1 ^= X0; }
    *o0 = X0 + ks2; *o1 = X1 + k0 + 5u;
}

__device__ float gumbel_at_flat(unsigned f) {
    const unsigned H = (unsigned)((unsigned long long)BB * VV / 2ull);
    unsigned x0, x1, o0, o1;
    if (f < H) { x0 = f; x1 = f + H; } else { x0 = f - H; x1 = f; }
    threefry2x32(0u, 42u, x0, x1, &o0, &o1);
    unsigned bits = (f < H) ? o0 : o1;
    float u = __uint_as_float(0x3f800000u | (bits >> 9)) - 1.0f;  // [0,1)
    u = fmaxf(u, 1.1754944e-38f);
    return -logf(-logf(u));
}

// Build a 128000-bit penalty bitmask in LDS from the 200 generated token ids.
__device__ __forceinline__ void build_mask(unsigned* mask, const int* toks, int n, int tid) {
    for (int w = tid; w < NMASKW; w += 256) mask[w] = 0u;
    __syncthreads();
    for (int i = tid; i < n; i += 256) {
        int t = toks[i];
        if (t >= 0 && t < VV) atomicOr(&mask[t >> 5], 1u << (t & 31));
    }
    __syncthreads();
}

// ---------------- K_zero: nontemporal zero-fill of probs region -------------
__global__ void k_zero_probs(float* __restrict__ out) {
    long long i = (long long)blockIdx.x * blockDim.x + threadIdx.x;   // float4 index
    const long long n4 = (long long)BB * VV / 4;
    if (i < n4) {
        f4 z = {0.f, 0.f, 0.f, 0.f};
        __builtin_nontemporal_store(z, (f4*)(out) + i);
    }
}

// ---------------- K_hist: per-row histogram -> top-50 threshold -------------
// One block (256 thr, 8 waves) per row. Streams the row through LDS with
// double-buffered async b128 loads; lane-private staging slots => no barriers
// between issue/consume, only s_wait_asynccnt. Penalty via LDS bitmask.
__global__ void k_hist(const float* __restrict__ logits,
                       const int* __restrict__ toks, int ntok,
                       float* __restrict__ thresh) {
    __shared__ unsigned bins[NBINS];
    __shared__ unsigned mask[NMASKW];
    __shared__ f4 stage[2][256];
    const int tid = threadIdx.x;
    const int r = blockIdx.x;

    for (int b = tid; b < NBINS; b += 256) bins[b] = 0u;
    build_mask(mask, toks, ntok, tid);   // includes __syncthreads

    const float* rowp = logits + (long long)r * VV;
    const unsigned stage_base = (unsigned)(unsigned long long)(void*)&stage[0][0];
    const int NCHUNK = VV / 1024;   // 125, exact

    async_ld_b128(stage_base + (unsigned)tid * 16u,
                  (unsigned long long)(rowp + tid * 4));
    for (int c = 0; c < NCHUNK; c++) {
        const int buf = c & 1;
        if (c + 1 < NCHUNK) {
            async_ld_b128(stage_base + (unsigned)((buf ^ 1) * 4096 + tid * 16),
                          (unsigned long long)(rowp + (c + 1) * 1024 + tid * 4));
            wait_async_1();       // oldest (chunk c) complete
        } else {
            wait_async_0();
        }
        f4 v = stage[buf][tid];
        const int j0 = c * 1024 + tid * 4;                 // 4 bits in one word
        const unsigned mw = mask[j0 >> 5] >> (j0 & 31);
        float xs[4] = { v.x * INV_T * ((mw      & 1u) ? INV_REP : 1.0f),
                        v.y * INV_T * ((mw >> 1 & 1u) ? INV_REP : 1.0f),
                        v.z * INV_T * ((mw >> 2 & 1u) ? INV_REP : 1.0f),
                        v.w * INV_T * ((mw >> 3 & 1u) ? INV_REP : 1.0f) };
        #pragma unroll
        for (int k = 0; k < 4; k++) {
            int b = (int)((xs[k] - HLO) * BINSCALE);
            b = b < 0 ? 0 : (b > NBINS - 1 ? NBINS - 1 : b);
            atomicAdd(&bins[b], 1u);
        }
    }
    __syncthreads();
    if (tid == 0) {
        int acc = 0; float t = HLO;
        for (int b = NBINS - 1; b >= 0; b--) {
            acc += (int)bins[b];
            if (acc >= 50) { t = HLO + (float)b * (1.0f / BINSCALE); break; }
        }
        thresh[r] = t;
    }
}

// ---------------- K_select: collect candidates, sort, finalize row ----------
__global__ void k_select(const float* __restrict__ logits,
                         const int* __restrict__ toks, int ntok,
                         const float* __restrict__ thresh,
                         int* __restrict__ keptCnt, int* __restrict__ keptIdx,
                         float* __restrict__ keptProb, float* __restrict__ entW,
                         int* __restrict__ tokW) {
    __shared__ float cVal[MAXC];
    __shared__ int   cIdx[MAXC];
    __shared__ unsigned mask[NMASKW];
    __shared__ float sVal[64];
    __shared__ int   sIdx[64];
    __shared__ int   cnt;
    const int tid = threadIdx.x;
    const int r = blockIdx.x;
    if (tid == 0) cnt = 0;
    build_mask(mask, toks, ntok, tid);

    const float thr = thresh[r];
    const float* rowp = logits + (long long)r * VV;
    const int NCHUNK = VV / 1024;
    for (int c = 0; c < NCHUNK; c++) {
        if (c + 4 < NCHUNK)
            __builtin_prefetch(rowp + (c + 4) * 1024 + tid * 4, 0, 0);  // global_prefetch_b8
        f4 v = *(const f4*)(rowp + c * 1024 + tid * 4);    // L2-resident re-read
        const int j0 = c * 1024 + tid * 4;
        const unsigned mw = mask[j0 >> 5] >> (j0 & 31);
        float xs[4] = { v.x * INV_T * ((mw      & 1u) ? INV_REP : 1.0f),
                        v.y * INV_T * ((mw >> 1 & 1u) ? INV_REP : 1.0f),
                        v.z * INV_T * ((mw >> 2 & 1u) ? INV_REP : 1.0f),
                        v.w * INV_T * ((mw >> 3 & 1u) ? INV_REP : 1.0f) };
        #pragma unroll
        for (int k = 0; k < 4; k++) {
            if (xs[k] >= thr) {
                int pos = atomicAdd(&cnt, 1);
                if (pos < MAXC) { cVal[pos] = xs[k]; cIdx[pos] = j0 + k; }
            }
        }
    }
    __syncthreads();
    const int C = cnt < MAXC ? cnt : MAXC;
    // exact rank (descending value, stable by index) -> top-64 into sVal/sIdx
    for (int i = tid; i < C; i += 256) {
        float vi = cVal[i]; int ii = cIdx[i]; int rank = 0;
        for (int j = 0; j < C; j++) {
            float vj = cVal[j];
            rank += (vj > vi) || (vj == vi && cIdx[j] < ii);
        }
        if (rank < 64) { sVal[rank] = vi; sIdx[rank] = ii; }
    }
    __syncthreads();

    if (tid == 0) {
        const int n50 = C < 50 ? C : 50;
        float e[50], cum[50];
        float m = sVal[0], denom = 0.f;
        for (int i = 0; i < n50; i++) { e[i] = expf(sVal[i] - m); denom += e[i]; }
        float run = 0.f;
        for (int i = 0; i < n50; i++) { run += e[i] / denom; cum[i] = run; }
        int K = 0;
        for (int i = 0; i < n50; i++) if (cum[i] <= TOPP) K++;
        int last = K - 1;
        if (last > 0 && cum[last - 1] >= TOPP * 0.98f) last--;
        int idx = (K == 0) ? 1 : (last + 1 < 5 ? last + 1 : 5);
        if (idx > n50) idx = n50;
        const float sel = cum[idx - 1];
        const float log_scale = logf(TOPP / sel);

        float ent = 0.f, best = -1e38f; int btok = sIdx[0];
        for (int i = 0; i < idx; i++) {
            float q = (e[i] / denom) / sel;           // == softmax over kept tokens
            ent -= q * logf(q + 1e-10f);
            float fl = sVal[i] + log_scale;           // filtered logit
            unsigned flat = (unsigned)r * (unsigned)VV + (unsigned)sIdx[i];
            float sc = fl + gumbel_at_flat(flat);     // Gumbel-max == categorical
            if (sc > best) { best = sc; btok = sIdx[i]; }
            keptIdx[r * 5 + i] = sIdx[i];
            keptProb[r * 5 + i] = q;
        }
        keptCnt[r] = idx; entW[r] = ent; tokW[r] = btok;
    }
}

// ---------------- K_scatter: write sparse probs + tokens + entropy ----------
__global__ void k_scatter(float* __restrict__ out,
                          const int* __restrict__ keptCnt, const int* __restrict__ keptIdx,
                          const float* __restrict__ keptProb, const float* __restrict__ entW,
                          const int* __restrict__ tokW) {
    int r = blockIdx.x * blockDim.x + threadIdx.x;
    if (r >= BB) return;
    int n = keptCnt[r];
    for (int i = 0; i < n; i++)
        out[(long long)r * VV + keptIdx[r * 5 + i]] = keptProb[r * 5 + i];
    out[(long long)BB * VV + r] = (float)tokW[r];          // next_tokens
    out[(long long)BB * VV + BB + r] = entW[r];            // entropy
}

extern "C" void kernel_launch(void* const* d_in, const int* in_sizes, int n_in,
                              void* d_out, int out_size, void* d_ws, size_t ws_size,
                              hipStream_t stream) {
    const float* logits = (const float*)d_in[0];
    const int* gtoks = (const int*)d_in[1];
    const int ntok = in_sizes[1];
    float* out = (float*)d_out;

    // workspace carve-up (penalty now lives in LDS as a bitmask; no pen array)
    float* thresh = (float*)d_ws;                 // BB
    int* keptCnt = (int*)(thresh + BB);           // BB
    int* keptIdx = keptCnt + BB;                  // BB*5
    float* keptProb = (float*)(keptIdx + BB * 5); // BB*5
    float* entW = keptProb + BB * 5;              // BB
    int* tokW = (int*)(entW + BB);                // BB

    // 1) zero-fill probs (131 MB, nontemporal streaming stores)
    const long long n4 = (long long)BB * VV / 4;
    k_zero_probs<<<(unsigned)((n4 + 255) / 256), 256, 0, stream>>>(out);
    // 2) per-row histogram -> top-50 threshold (async global->LDS streaming)
    k_hist<<<BB, 256, 0, stream>>>(logits, gtoks, ntok, thresh);
    // 3) candidate collect + exact top-50 + top-p finalize + Gumbel sample
    k_select<<<BB, 256, 0, stream>>>(logits, gtoks, ntok, thresh,
                                     keptCnt, keptIdx, keptProb, entW, tokW);
    // 4) scatter sparse results
    k_scatter<<<1, 256, 0, stream>>>(out, keptCnt, keptIdx, keptProb, entW, tokW);
}